// GraphSAGE_82042465288656
// MI455X (gfx1250) — compile-verified
//
#include <hip/hip_runtime.h>
#include <stdint.h>

typedef float v2f __attribute__((ext_vector_type(2)));
typedef float v8f __attribute__((ext_vector_type(8)));
typedef unsigned int v4u __attribute__((ext_vector_type(4)));
typedef int v4i __attribute__((ext_vector_type(4)));
typedef int v8i __attribute__((ext_vector_type(8)));

#define N_NODES 50000
#define N_EDGES 800000
#define DIM 256
#define LAYERS 3
#define ROW_TILES (N_NODES / 16)   // 3125
#define GEMM_BLKX 50               // row-group blocks per column tile

// ---------------- utility kernels ----------------

__global__ void zero_f32(float* __restrict__ p, int n) {
  int i = blockIdx.x * blockDim.x + threadIdx.x;
  if (i < n) p[i] = 0.0f;
}

__global__ void degree_kernel(const int* __restrict__ tgt, float* __restrict__ deg) {
  int e = blockIdx.x * blockDim.x + threadIdx.x;
  if (e < N_EDGES) atomicAdd(&deg[tgt[e]], 1.0f);
}

__global__ void invdeg_kernel(float* __restrict__ deg) {
  int n = blockIdx.x * blockDim.x + threadIdx.x;
  if (n < N_NODES) {
    float d = deg[n];
    deg[n] = (d > 0.0f) ? (1.0f / d) : 0.0f;
  }
}

// ---------------- edge scatter (mean-agg numerator) ----------------
__global__ void scatter_kernel(const int* __restrict__ src, const int* __restrict__ tgt,
                               const float* __restrict__ x, int strideX, int offX,
                               float* __restrict__ agg) {
  long long t = (long long)blockIdx.x * blockDim.x + threadIdx.x;
  int e = (int)(t >> 6);           // edge id
  int d = ((int)t & 63) << 2;      // feature chunk start
  if (e >= N_EDGES) return;
  int s = src[e];
  int g = tgt[e];
  const float4 v = *(const float4*)(x + (long long)s * strideX + offX + d);
  float* o = agg + (long long)g * DIM + d;
  atomicAdd(o + 0, v.x);
  atomicAdd(o + 1, v.y);
  atomicAdd(o + 2, v.z);
  atomicAdd(o + 3, v.w);
}

// ---------------- TDM: DMA a 256x16 f32 column panel of W into LDS ----------------
// D# built per CDNA5 ISA: 2-D tensor, data_size=4B,
// tensor_dim0=256 (row length), stride0=256, tile_dim0=16, tile_dim1=256.
// Panel lands row-major in LDS: panel[k][0..15] contiguous per k.
__device__ __forceinline__ void tdm_load_panel_f32(const void* gptr, unsigned lds_bytes) {
  unsigned long long ga = (unsigned long long)(uintptr_t)gptr;
  v4u g0;
  g0.x = 1u;                                                // count=1, user descriptor
  g0.y = lds_bytes;                                         // lds_addr (bytes)
  g0.z = (unsigned)(ga & 0xFFFFFFFFu);                      // global_addr[31:0]
  g0.w = (unsigned)((ga >> 32) & 0x01FFFFFFu) | (2u << 30); // addr[56:32] | type=2
  v8i g1;
  g1[0] = (int)(2u << 16);       // workgroup_mask=0, data_size=2 (4 bytes)
  g1[1] = (int)(256u << 16);     // tensor_dim0[15:0] = 256
  g1[2] = (int)(256u << 16);     // tensor_dim0[31:16]=0 | tensor_dim1[15:0]=256
  g1[3] = (int)(16u << 16);      // tensor_dim1[31:16]=0 | tile_dim0=16
  g1[4] = 256;                   // tile_dim1=256, tile_dim2=0
  g1[5] = 256;                   // tensor_dim0_stride[31:0] = 256
  g1[6] = 0;                     // stride0 hi | tensor_dim1_stride lo (unused, 2-D)
  g1[7] = 0;
  v4i z4 = {};
#if __clang_major__ >= 23
  v8i z8 = {};
  __builtin_amdgcn_tensor_load_to_lds(g0, g1, z4, z4, z8, 0);
#else
  __builtin_amdgcn_tensor_load_to_lds(g0, g1, z4, z4, 0);
#endif
}

// ---------------- fused SAGE layer GEMM via fp32 WMMA + TDM-staged B ----------------
// out[:, outOff:outOff+256] = act( (agg*inv_deg) @ Wl + x @ Wr + bias + x )
// Block: one 16-col tile; TDM stages Wl/Wr panels into LDS once; 8 waves
// grid-stride the 3125 row tiles, issuing V_WMMA_F32_16X16X4_F32 pairs with
// A from global (L2-resident) and B from LDS.
__global__ void __launch_bounds__(256)
sage_gemm_kernel(const float* __restrict__ agg, const float* __restrict__ inv_deg,
                 const float* __restrict__ xin, int strideX, int offX,
                 const float* __restrict__ Wl, const float* __restrict__ Wr,
                 const float* __restrict__ bias, float* __restrict__ out,
                 int outOff, int do_relu) {
  __shared__ float ldsW[2 * DIM * 16];   // [0:4096)=Wl panel, [4096:8192)=Wr panel

  const int col0 = (int)blockIdx.y << 4;
  const int wid  = threadIdx.x >> 5;
  const int lane = threadIdx.x & 31;
  const int half = lane >> 4;            // 0: K pair {0,1}, 1: K pair {2,3}
  const int l    = lane & 15;

  if (threadIdx.x < 32) {                // wave 0 drives the DMA engine
    unsigned base = (unsigned)(uintptr_t)&ldsW[0];
    tdm_load_panel_f32(Wl + col0, base);
    tdm_load_panel_f32(Wr + col0, base + DIM * 16 * sizeof(float));
    __builtin_amdgcn_s_wait_tensorcnt(0);
  }
  __syncthreads();                       // publish LDS panels to all 8 waves

  const int   col = col0 + l;
  const float bv  = bias[col];           // per-lane output-column bias

  for (int rt = (int)blockIdx.x * 8 + wid; rt < ROW_TILES; rt += GEMM_BLKX * 8) {
    const int row0 = rt << 4;
    const float inv = inv_deg[row0 + l];
    const float* aggRow = agg + (long long)(row0 + l) * DIM;
    const float* xRow   = xin + (long long)(row0 + l) * strideX + offX;

    v8f c = {};
#pragma unroll 4
    for (int k0 = 0; k0 < DIM; k0 += 4) {
      const int k = k0 + (half << 1);
      const float2 ta = *(const float2*)(aggRow + k);
      const float2 tx = *(const float2*)(xRow + k);
      v2f aL; aL.x = ta.x * inv; aL.y = ta.y * inv;    // fold mean normalizer
      v2f aR; aR.x = tx.x;       aR.y = tx.y;
      v2f bL; bL.x = ldsW[k * 16 + l];
              bL.y = ldsW[(k + 1) * 16 + l];
      v2f bR; bR.x = ldsW[DIM * 16 + k * 16 + l];
              bR.y = ldsW[DIM * 16 + (k + 1) * 16 + l];

      c = __builtin_amdgcn_wmma_f32_16x16x4_f32(false, aL, false, bL,
                                                (short)0, c, false, false);
      c = __builtin_amdgcn_wmma_f32_16x16x4_f32(false, aR, false, bR,
                                                (short)0, c, false, false);
    }

    // Epilogue: bias + residual (+ReLU). C layout: VGPR v -> rows
    // row0 + 8*half + v (consecutive), so walk two pointers instead of
    // re-multiplying row*stride per element.
    const int rbase = row0 + (half << 3);
    const float* resp = xin + (long long)rbase * strideX + offX + col;
    float*       outp = out + (long long)rbase * (3 * DIM) + outOff + col;
#pragma unroll
    for (int v = 0; v < 8; ++v) {
      float val = c[v] + bv + *resp;
      if (do_relu) val = fmaxf(val, 0.0f);
      *outp = val;
      resp += strideX;
      outp += 3 * DIM;
    }
  }
}

// ---------------- launcher ----------------

extern "C" void kernel_launch(void* const* d_in, const int* in_sizes, int n_in,
                              void* d_out, int out_size, void* d_ws, size_t ws_size,
                              hipStream_t stream) {
  const int*   edge  = (const int*)d_in[0];    // (2, E)
  const float* x0    = (const float*)d_in[1];  // (N, D)
  const float* W_l   = (const float*)d_in[2];  // (L, D, D)
  const float* W_r   = (const float*)d_in[3];  // (L, D, D)
  const float* bias  = (const float*)d_in[4];  // (L, D)
  float*       out   = (float*)d_out;          // (N, 3*D)

  const int* src = edge;
  const int* tgt = edge + N_EDGES;

  float* inv_deg = (float*)d_ws;                       // N floats
  float* agg     = inv_deg + ((N_NODES + 255) & ~255); // N*D floats, aligned

  // Degree -> inv_deg (computed once per call)
  zero_f32<<<(N_NODES + 255) / 256, 256, 0, stream>>>(inv_deg, N_NODES);
  degree_kernel<<<(N_EDGES + 255) / 256, 256, 0, stream>>>(tgt, inv_deg);
  invdeg_kernel<<<(N_NODES + 255) / 256, 256, 0, stream>>>(inv_deg);

  for (int i = 0; i < LAYERS; ++i) {
    const float* xin;
    int strideX, offX;
    if (i == 0) { xin = x0;  strideX = DIM;     offX = 0; }
    else        { xin = out; strideX = 3 * DIM; offX = (i - 1) * DIM; }

    zero_f32<<<(N_NODES * DIM + 255) / 256, 256, 0, stream>>>(agg, N_NODES * DIM);

    const long long chunks = (long long)N_EDGES * 64;
    scatter_kernel<<<(int)((chunks + 255) / 256), 256, 0, stream>>>(
        src, tgt, xin, strideX, offX, agg);

    // 16 column tiles x 50 row-group blocks, 8 waves/block
    sage_gemm_kernel<<<dim3(GEMM_BLKX, 16), 256, 0, stream>>>(
        agg, inv_deg, xin, strideX, offX,
        W_l + (long long)i * DIM * DIM, W_r + (long long)i * DIM * DIM,
        bias + i * DIM, out, i * DIM, (i < LAYERS - 1) ? 1 : 0);
  }
}